// nconv_11338713661504
// MI455X (gfx1250) — compile-verified
//
#include <hip/hip_runtime.h>

// Quaternion graph-conv: out = Hamilton(A (x) x), fused 4-GEMM with
// V_WMMA_F32_16X16X4_F32. 128x128 block tiles, 16 waves (4x4 grid).
// Double-buffered K tiles staged with GLOBAL_LOAD_ASYNC_TO_LDS (ASYNCcnt),
// overlapping the DMA of tile t+1 with the 256 WMMAs of tile t.

typedef __attribute__((ext_vector_type(2))) float v2f;
typedef __attribute__((ext_vector_type(8))) float v8f;

#define BQ 16
#define CQ 128
#define NQ 4096
#define TQ 12
#define FQ 32                   // C/4 channels per quaternion component
#define JCOLS (BQ * FQ * TQ)    // 6144 columns per quadrant
#define NB 128                  // rows per block
#define JB 128                  // cols per block
#define KT 16                   // K tile per buffer
#define APITCH 20               // floats: 80B rows, 16B aligned, conflict-free
#define XPITCH 18               // floats: conflict-free b64 fragment reads
#define AC (128 * APITCH)       // 2560 floats per A component tile
#define XC (128 * XPITCH)       // 2304 floats per X quadrant tile
#define ATILE (4 * AC)          // 10240 floats
#define XTILE (4 * XC)          // 9216 floats
#define BUFF (ATILE + XTILE)    // 19456 floats per buffer

static __device__ __forceinline__ v8f wmma4(v2f a, v2f b, v8f c) {
  return __builtin_amdgcn_wmma_f32_16x16x4_f32(false, a, false, b, (short)0, c,
                                               false, false);
}

// Async global->LDS DMA (per-lane). ldsoff = wave-relative LDS byte address,
// goff = 32-bit byte offset added to the uniform 64-bit base (GVS mode).
static __device__ __forceinline__ void async_ld_b128(unsigned ldsoff,
                                                     const float* base,
                                                     unsigned goff) {
  asm volatile("global_load_async_to_lds_b128 %0, %1, %2"
               :: "v"(ldsoff), "v"(goff), "s"(base) : "memory");
}
static __device__ __forceinline__ void async_ld_b32(unsigned ldsoff,
                                                    const float* base,
                                                    unsigned goff) {
  asm volatile("global_load_async_to_lds_b32 %0, %1, %2"
               :: "v"(ldsoff), "v"(goff), "s"(base) : "memory");
}
static __device__ __forceinline__ void wait_async0() {
  asm volatile("s_wait_asynccnt 0x0" ::: "memory");
}

__global__ __launch_bounds__(512) void quat_gconv_wmma3(
    const float* __restrict__ x,
    const float* __restrict__ Ar, const float* __restrict__ Ai,
    const float* __restrict__ Aj, const float* __restrict__ Ak,
    float* __restrict__ out) {
  __shared__ float smem[2 * BUFF];  // 155,648 B: two (A,X) K-tile buffers

  const int tid  = threadIdx.x;
  const int lane = tid & 31;
  const int wave = tid >> 5;
  const int wrow = wave >> 2;  // 0..3
  const int wcol = wave & 3;   // 0..3

  const int n0 = blockIdx.x * NB;
  const int jb = blockIdx.y * JB;

  const int lcol  = lane & 15;
  const int khalf = (lane >> 4) << 1;  // K sub-offset 0 or 2

  // Low 32 bits of a generic LDS pointer == wave-relative LDS byte address.
  const unsigned smem_base = (unsigned)(unsigned long long)(void*)&smem[0];

  // output column bases: 2 col-subtiles x 4 quadrants
  int obase[2][4];
#pragma unroll
  for (int cs = 0; cs < 2; ++cs) {
    const int j  = jb + wcol * 32 + cs * 16 + lcol;
    const int b  = j / (FQ * TQ);
    const int ft = j % (FQ * TQ);
    const int f  = ft / TQ;
    const int t  = ft % TQ;
#pragma unroll
    for (int p = 0; p < 4; ++p)
      obase[cs][p] = ((b * CQ + p * FQ + f) * NQ) * TQ + t;
  }

  // X staging mapping: each thread owns one column, walks m
  const int scol  = tid & 127;
  const int smrow = tid >> 7;  // 0..3
  int sxbase[4];
  {
    const int j  = jb + scol;
    const int b  = j / (FQ * TQ);
    const int ft = j % (FQ * TQ);
    const int f  = ft / TQ;
    const int t  = ft % TQ;
#pragma unroll
    for (int p = 0; p < 4; ++p)
      sxbase[p] = ((b * CQ + p * FQ + f) * NQ) * TQ + t;
  }

  // A staging mapping: one float4 per comp per thread
  const int arow = tid >> 2;  // 0..127
  const int ac4  = tid & 3;   // float4 within 16-wide K row

  const float* Acomp[4] = {Ar, Ai, Aj, Ak};

  v8f acc[2][2][4];  // [rowsub][colsub][quadrant]
#pragma unroll
  for (int rs = 0; rs < 2; ++rs)
#pragma unroll
    for (int cs = 0; cs < 2; ++cs)
#pragma unroll
      for (int p = 0; p < 4; ++p)
        acc[rs][cs][p] = (v8f){0.f, 0.f, 0.f, 0.f, 0.f, 0.f, 0.f, 0.f};

  // ---- async stage of one K tile into buffer `buf` ----
  auto stage_tile = [&](int k0, int buf) {
    const unsigned aoff = smem_base + (unsigned)(buf * BUFF) * 4u;
    const unsigned xoff = aoff + (unsigned)ATILE * 4u;
#pragma unroll
    for (int c = 0; c < 4; ++c) {
      const unsigned l = aoff + (unsigned)(c * AC + arow * APITCH + ac4 * 4) * 4u;
      const unsigned g = (unsigned)(((n0 + arow) * NQ + k0 + ac4 * 4) * 4);
      async_ld_b128(l, Acomp[c], g);
    }
#pragma unroll
    for (int p = 0; p < 4; ++p) {
#pragma unroll
      for (int it = 0; it < 4; ++it) {
        const int m = smrow + it * 4;  // 0..15
        const unsigned l = xoff + (unsigned)(p * XC + scol * XPITCH + m) * 4u;
        const unsigned g = (unsigned)((sxbase[p] + (k0 + m) * TQ) * 4);
        async_ld_b32(l, x, g);
      }
    }
  };

  // prologue: fill buffer 0
  stage_tile(0, 0);
  wait_async0();
  __syncthreads();

  for (int k0 = 0; k0 < NQ; k0 += KT) {
    const int pbuf = (k0 / KT) & 1;
    if (k0 + KT < NQ)             // uniform branch: EXEC stays all-ones
      stage_tile(k0 + KT, pbuf ^ 1);

    const float* Ab = smem + pbuf * BUFF;
    const float* Xb = Ab + ATILE;
    const int ab0 = (wrow * 32 + lcol) * APITCH + khalf;
    const int xb0 = (wcol * 32 + lcol) * XPITCH + khalf;

#pragma unroll 2
    for (int kk = 0; kk < KT; kk += 4) {
      v2f a[2][4], xf[2][4];
#pragma unroll
      for (int rs = 0; rs < 2; ++rs)
#pragma unroll
        for (int c = 0; c < 4; ++c)
          a[rs][c] = *(const v2f*)&Ab[c * AC + ab0 + rs * (16 * APITCH) + kk];
#pragma unroll
      for (int cs = 0; cs < 2; ++cs)
#pragma unroll
        for (int p = 0; p < 4; ++p)
          xf[cs][p] = *(const v2f*)&Xb[p * XC + xb0 + cs * (16 * XPITCH) + kk];

#pragma unroll
      for (int cs = 0; cs < 2; ++cs) {
        const v2f xr = xf[cs][0], xi = xf[cs][1];
        const v2f xj = xf[cs][2], xk = xf[cs][3];
        const v2f nxi = -xi, nxj = -xj, nxk = -xk;
#pragma unroll
        for (int rs = 0; rs < 2; ++rs) {
          const v2f ar = a[rs][0], ai = a[rs][1];
          const v2f aj = a[rs][2], ak = a[rs][3];
          // Hamilton product: 16 WMMAs into 4 quadrant accumulators
          acc[rs][cs][0] = wmma4(ar, xr,  acc[rs][cs][0]);
          acc[rs][cs][0] = wmma4(ai, nxi, acc[rs][cs][0]);
          acc[rs][cs][0] = wmma4(aj, nxj, acc[rs][cs][0]);
          acc[rs][cs][0] = wmma4(ak, nxk, acc[rs][cs][0]);

          acc[rs][cs][1] = wmma4(ar, xi,  acc[rs][cs][1]);
          acc[rs][cs][1] = wmma4(ai, xr,  acc[rs][cs][1]);
          acc[rs][cs][1] = wmma4(aj, xk,  acc[rs][cs][1]);
          acc[rs][cs][1] = wmma4(ak, nxj, acc[rs][cs][1]);

          acc[rs][cs][2] = wmma4(ar, xj,  acc[rs][cs][2]);
          acc[rs][cs][2] = wmma4(ai, nxk, acc[rs][cs][2]);
          acc[rs][cs][2] = wmma4(aj, xr,  acc[rs][cs][2]);
          acc[rs][cs][2] = wmma4(ak, xi,  acc[rs][cs][2]);

          acc[rs][cs][3] = wmma4(ar, xk,  acc[rs][cs][3]);
          acc[rs][cs][3] = wmma4(ai, xj,  acc[rs][cs][3]);
          acc[rs][cs][3] = wmma4(aj, nxi, acc[rs][cs][3]);
          acc[rs][cs][3] = wmma4(ak, xr,  acc[rs][cs][3]);
        }
      }
    }

    // this wave's DMA for tile t+1 done, then block-wide: safe to flip buffers
    wait_async0();
    __syncthreads();
  }

  // Epilogue. C/D layout: VGPR r holds row M = r + 8*(lane>>4), col = lane&15
  const int mhi = (lane >> 4) << 3;
#pragma unroll
  for (int rs = 0; rs < 2; ++rs) {
#pragma unroll
    for (int r = 0; r < 8; ++r) {
      const int noff = (n0 + wrow * 32 + rs * 16 + r + mhi) * TQ;
#pragma unroll
      for (int cs = 0; cs < 2; ++cs) {
#pragma unroll
        for (int p = 0; p < 4; ++p)
          out[obase[cs][p] + noff] = acc[rs][cs][p][r];
      }
    }
  }
}

extern "C" void kernel_launch(void* const* d_in, const int* in_sizes, int n_in,
                              void* d_out, int out_size, void* d_ws,
                              size_t ws_size, hipStream_t stream) {
  const float* x  = (const float*)d_in[0];
  const float* Ar = (const float*)d_in[1];
  const float* Ai = (const float*)d_in[2];
  const float* Aj = (const float*)d_in[3];
  const float* Ak = (const float*)d_in[4];
  float* out = (float*)d_out;

  dim3 grid(NQ / NB, JCOLS / JB);  // 32 x 48 blocks
  dim3 block(512);                 // 16 wave32, 4x4 wave grid
  quat_gconv_wmma3<<<grid, block, 0, stream>>>(x, Ar, Ai, Aj, Ak, out);
}